// Bottleneck2_32504312496475
// MI455X (gfx1250) — compile-verified
//
#include <hip/hip_runtime.h>
#include <hip/hip_bf16.h>

// ---------------------------------------------------------------------------
// CDNA5 (gfx1250) implementation of the deformable-conv bottleneck.
// All GEMM-shaped ops use V_WMMA_F32_16X16X4_F32 (native fp32 WMMA, wave32).
// Workspace requirement: ~91 MB (see layout below).
// ---------------------------------------------------------------------------

typedef float v2f __attribute__((ext_vector_type(2)));
typedef float v8f __attribute__((ext_vector_type(8)));

#define B_     8
#define CIN    256
#define P_     128
#define COUT   512
#define H_     56
#define W_     56
#define HW     3136          // 56*56 = 49 * 64
#define HP     58
#define HPWP   3364          // 58*58
#define KDC    1152          // 128 * 9
#define CNT_   25088.0f      // B*H*W
#define EPS_   1e-5f

// ---------------------------------------------------------------------------
// Small utility kernels
// ---------------------------------------------------------------------------
__global__ void zero_f(float* p, int n) {
    int i = blockIdx.x * blockDim.x + threadIdx.x;
    if (i < n) p[i] = 0.f;
}

__global__ void finalize_bn(const float* __restrict__ sum, const float* __restrict__ sq,
                            const float* __restrict__ g, const float* __restrict__ bt,
                            float* __restrict__ sc, float* __restrict__ bi, int nch) {
    int i = blockIdx.x * blockDim.x + threadIdx.x;
    if (i >= nch) return;
    float mean = sum[i] / CNT_;
    float var  = sq[i] / CNT_ - mean * mean;
    float s    = g[i] * rsqrtf(var + EPS_);
    sc[i] = s;
    bi[i] = bt[i] - mean * s;
}

// xp[b,c,1+h,1+w] = relu(a1[c]*y1[b,c,h,w] + c1[c]);  zero border (PAD=1)
__global__ void build_xp(const float* __restrict__ y1, const float* __restrict__ a1,
                         const float* __restrict__ c1, float* __restrict__ xp) {
    int t  = blockIdx.x * blockDim.x + threadIdx.x;   // exactly B*P*HPWP threads
    int wp = t % HP;
    int hp = (t / HP) % HP;
    int c  = (t / HPWP) % P_;
    int b  = t / (P_ * HPWP);
    float v = 0.f;
    if (hp >= 1 && hp <= H_ && wp >= 1 && wp <= W_) {
        float y = y1[((size_t)(b * P_ + c)) * HW + (size_t)(hp - 1) * W_ + (wp - 1)];
        v = fmaxf(a1[c] * y + c1[c], 0.f);
    }
    xp[t] = v;
}

// wdcT[o, n*128 + c] = w_dc[o, c*9 + n]   (K reorder so taps are outer)
__global__ void transpose_wdc(const float* __restrict__ w_dc, float* __restrict__ wdcT) {
    int i = blockIdx.x * blockDim.x + threadIdx.x;    // exactly 128*1152 threads
    int o = i / KDC;
    int r = i % KDC;
    int n = r / P_;
    int c = r % P_;
    wdcT[(size_t)o * KDC + n * P_ + c] = w_dc[(size_t)o * KDC + c * 9 + n];
}

// ---------------------------------------------------------------------------
// Offset conv (3x3, 18 ch) + bilinear table build. One thread per output pixel.
// w_off (18*128*9 = 20736 floats = 83KB) staged in LDS (320KB/WGP available).
// ---------------------------------------------------------------------------
__global__ __launch_bounds__(128)
void offsets_tables(const float* __restrict__ xp, const float* __restrict__ w_off,
                    const float* __restrict__ b_off,
                    float4* __restrict__ tblW, int4* __restrict__ tblI) {
    __shared__ float sw[18 * KDC];   // [ (p*9+tap)*18 + o ]
    for (int i = threadIdx.x; i < 18 * KDC; i += 128) {
        int o = i / KDC;           // w_off flat layout: o*1152 + p*9 + tap
        int r = i % KDC;
        sw[r * 18 + o] = w_off[i];
    }
    __syncthreads();

    int t   = blockIdx.x * 128 + threadIdx.x;   // exactly B*HW threads (196 blocks)
    int b   = t / HW;
    int pix = t % HW;
    int h   = pix / W_;
    int w   = pix % W_;

    float acc[18];
#pragma unroll
    for (int o = 0; o < 18; ++o) acc[o] = b_off[o];

    const float* xpb = xp + (size_t)b * P_ * HPWP;
    for (int p = 0; p < P_; ++p) {
        const float* xr = xpb + (size_t)p * HPWP + h * HP + w;  // 3x3 window top-left
#pragma unroll
        for (int kh = 0; kh < 3; ++kh)
#pragma unroll
            for (int kw = 0; kw < 3; ++kw) {
                float v = xr[kh * HP + kw];
                const float* wrow = &sw[(p * 9 + kh * 3 + kw) * 18];
#pragma unroll
                for (int o = 0; o < 18; ++o) acc[o] += wrow[o] * v;
            }
    }

    // Bilinear sampling tables (reference semantics, stop_gradient == identity fwd)
#pragma unroll
    for (int n = 0; n < 9; ++n) {
        float px = acc[2 * n]     + (float)(n / 3 - 1) + (float)(h + 1);
        float py = acc[2 * n + 1] + (float)(n % 3 - 1) + (float)(w + 1);
        float fx = floorf(px), fy = floorf(py);
        float qltx = fminf(fmaxf(fx,       0.f), 57.f);
        float qlty = fminf(fmaxf(fy,       0.f), 57.f);
        float qrbx = fminf(fmaxf(fx + 1.f, 0.f), 57.f);
        float qrby = fminf(fmaxf(fy + 1.f, 0.f), 57.f);
        bool mx = (px < 1.f) || (px > 56.f);
        bool my = (py < 1.f) || (py > 56.f);
        px = mx ? fx : px;  py = my ? fy : py;
        px = fminf(fmaxf(px, 0.f), 57.f);
        py = fminf(fmaxf(py, 0.f), 57.f);
        float glt = (1.f + (qltx - px)) * (1.f + (qlty - py));
        float grb = (1.f - (qrbx - px)) * (1.f - (qrby - py));
        float glb = (1.f + (qltx - px)) * (1.f - (qrby - py));
        float grt = (1.f - (qrbx - px)) * (1.f + (qlty - py));
        int ilt = (int)qltx * HP + (int)qlty;
        int irb = (int)qrbx * HP + (int)qrby;
        int ilb = (int)qltx * HP + (int)qrby;
        int irt = (int)qrbx * HP + (int)qlty;
        size_t ti = ((size_t)b * HW + pix) * 9 + n;
        tblW[ti] = make_float4(glt, grb, glb, grt);
        tblI[ti] = make_int4(ilt, irb, ilb, irt);
    }
}

// ---------------------------------------------------------------------------
// Shared epilogue: store 16x64 wave tile + fused BN statistics
// (cross-lane reduce within each 16-lane half, then one atomic per channel).
// ---------------------------------------------------------------------------
__device__ __forceinline__
void store_tile_stats(float* __restrict__ Yb, const v8f* acc, int mbase, int nbase,
                      int lq, int half, float* __restrict__ sum, float* __restrict__ sq) {
#pragma unroll
    for (int j = 0; j < 4; ++j) {
        const int n = nbase + j * 16 + lq;
#pragma unroll
        for (int r = 0; r < 8; ++r) {
            const int mm = mbase + r + 8 * half;   // 32-bit C/D layout: M = r + 8*(lane/16)
            float v = acc[j][r];
            Yb[(size_t)mm * HW + n] = v;
            float s = v, s2 = v * v;
#pragma unroll
            for (int d = 8; d >= 1; d >>= 1) {     // reduce over the 16 N-lanes
                s  += __shfl_xor(s,  d, 32);
                s2 += __shfl_xor(s2, d, 32);
            }
            if (lq == 0) {
                atomicAdd(&sum[mm], s);
                atomicAdd(&sq[mm],  s2);
            }
        }
    }
}

// ---------------------------------------------------------------------------
// 1x1 conv as fp32 WMMA GEMM:  Y[b,m,n] = sum_k A[m,k] * f(B[b,k,n])
// f = identity (bsc==null) or relu(bsc[k]*x + bbi[k])  (fused BN+relu input).
// Block = 256 threads = 8 waves; wave tile = 16(M) x 64(N); grid (49, M/128, 8).
// ---------------------------------------------------------------------------
__global__ __launch_bounds__(256)
void gemm1x1_wmma(const float* __restrict__ A, const float* __restrict__ Bsrc,
                  float* __restrict__ Y, const float* __restrict__ bsc,
                  const float* __restrict__ bbi, int M, int K,
                  float* __restrict__ sum, float* __restrict__ sq) {
    const int lane  = threadIdx.x & 31;
    const int wave  = threadIdx.x >> 5;
    const int half  = lane >> 4;        // A/B 32-bit layout: K = k0 + 2*(lane/16) + v
    const int lq    = lane & 15;
    const int b     = blockIdx.z;
    const int nbase = blockIdx.x * 64;
    const int mbase = blockIdx.y * 128 + wave * 16;
    const float* Bb = Bsrc + (size_t)b * K * HW;
    float*       Yb = Y    + (size_t)b * M * HW;

    v8f acc[4] = {};
    const int m = mbase + lq;
    for (int k0 = 0; k0 < K; k0 += 4) {
        const int kk = k0 + 2 * half;
        v2f a = *(const v2f*)(A + (size_t)m * K + kk);      // kk even -> 8B aligned
        const float* r0 = Bb + (size_t)kk * HW + nbase + lq;
        const float* r1 = r0 + HW;
        float b00 = r0[0],  b01 = r1[0];
        float b10 = r0[16], b11 = r1[16];
        float b20 = r0[32], b21 = r1[32];
        float b30 = r0[48], b31 = r1[48];
        if (bsc) {                                          // fused BN+relu on input
            float s0 = bsc[kk], o0 = bbi[kk];
            float s1 = bsc[kk + 1], o1 = bbi[kk + 1];
            b00 = fmaxf(b00 * s0 + o0, 0.f);  b01 = fmaxf(b01 * s1 + o1, 0.f);
            b10 = fmaxf(b10 * s0 + o0, 0.f);  b11 = fmaxf(b11 * s1 + o1, 0.f);
            b20 = fmaxf(b20 * s0 + o0, 0.f);  b21 = fmaxf(b21 * s1 + o1, 0.f);
            b30 = fmaxf(b30 * s0 + o0, 0.f);  b31 = fmaxf(b31 * s1 + o1, 0.f);
        }
        v2f v0 = {b00, b01}, v1 = {b10, b11}, v2 = {b20, b21}, v3 = {b30, b31};
        acc[0] = __builtin_amdgcn_wmma_f32_16x16x4_f32(false, a, false, v0, (short)0, acc[0], false, false);
        acc[1] = __builtin_amdgcn_wmma_f32_16x16x4_f32(false, a, false, v1, (short)0, acc[1], false, false);
        acc[2] = __builtin_amdgcn_wmma_f32_16x16x4_f32(false, a, false, v2, (short)0, acc[2], false, false);
        acc[3] = __builtin_amdgcn_wmma_f32_16x16x4_f32(false, a, false, v3, (short)0, acc[3], false, false);
    }
    store_tile_stats(Yb, acc, mbase, nbase, lq, half, sum, sq);
}

// ---------------------------------------------------------------------------
// Deformable conv as fp32 WMMA GEMM: out2[b,o,pix] = sum_{n,c} wdcT[o,n*128+c] *
//   bilinear(xp[b,c,:], tbl[b,pix,n]).  Tables hoisted per tap (reused over 128 c).
// ---------------------------------------------------------------------------
__global__ __launch_bounds__(256)
void deform_wmma(const float* __restrict__ wdcT, const float* __restrict__ xp,
                 const float4* __restrict__ tblW, const int4* __restrict__ tblI,
                 float* __restrict__ out2, float* __restrict__ sum, float* __restrict__ sq) {
    const int lane  = threadIdx.x & 31;
    const int wave  = threadIdx.x >> 5;
    const int half  = lane >> 4;
    const int lq    = lane & 15;
    const int b     = blockIdx.z;
    const int nbase = blockIdx.x * 64;
    const int mbase = wave * 16;                 // 8 waves cover M = 128
    const float* xpb = xp + (size_t)b * P_ * HPWP;

    v8f acc[4] = {};
    const int m = mbase + lq;
    for (int n = 0; n < 9; ++n) {
        float4 w4[4]; int4 ix[4];
#pragma unroll
        for (int j = 0; j < 4; ++j) {            // per-pixel tap tables, reused over c
            size_t ti = ((size_t)b * HW + (nbase + j * 16 + lq)) * 9 + n;
            w4[j] = tblW[ti];
            ix[j] = tblI[ti];
        }
        for (int c0 = 0; c0 < P_; c0 += 4) {
            const int kk = c0 + 2 * half;
            v2f a = *(const v2f*)(wdcT + (size_t)m * KDC + n * P_ + kk);
            const float* x0 = xpb + (size_t)kk * HPWP;
            const float* x1 = x0 + HPWP;
#pragma unroll
            for (int j = 0; j < 4; ++j) {
                v2f bv;
                bv.x = w4[j].x * x0[ix[j].x] + w4[j].y * x0[ix[j].y]
                     + w4[j].z * x0[ix[j].z] + w4[j].w * x0[ix[j].w];
                bv.y = w4[j].x * x1[ix[j].x] + w4[j].y * x1[ix[j].y]
                     + w4[j].z * x1[ix[j].z] + w4[j].w * x1[ix[j].w];
                acc[j] = __builtin_amdgcn_wmma_f32_16x16x4_f32(false, a, false, bv,
                                                               (short)0, acc[j], false, false);
            }
        }
    }
    store_tile_stats(out2 + (size_t)b * P_ * HW, acc, mbase, nbase, lq, half, sum, sq);
}

// out = relu( a3*y3 + c3 + as*ys + cs ), where d_out currently holds ys.
__global__ void final_out(const float* __restrict__ y3, float* __restrict__ out,
                          const float* __restrict__ a3, const float* __restrict__ c3,
                          const float* __restrict__ as_, const float* __restrict__ cs_) {
    size_t i = (size_t)blockIdx.x * blockDim.x + threadIdx.x;  // exactly B*COUT*HW
    int ch = (int)((i / HW) % COUT);
    float v = a3[ch] * y3[i] + c3[ch] + as_[ch] * out[i] + cs_[ch];
    out[i] = fmaxf(v, 0.f);
}

// ---------------------------------------------------------------------------
// Host-side orchestration (all launches on `stream`; graph-capture safe).
// ---------------------------------------------------------------------------
extern "C" void kernel_launch(void* const* d_in, const int* in_sizes, int n_in,
                              void* d_out, int out_size, void* d_ws, size_t ws_size,
                              hipStream_t stream) {
    const float* x      = (const float*)d_in[0];
    const float* w1     = (const float*)d_in[1];
    const float* g1     = (const float*)d_in[2];
    const float* b1     = (const float*)d_in[3];
    const float* w_off  = (const float*)d_in[4];
    const float* b_off  = (const float*)d_in[5];
    const float* w_dc   = (const float*)d_in[6];
    const float* g2     = (const float*)d_in[7];
    const float* b2     = (const float*)d_in[8];
    const float* w3     = (const float*)d_in[9];
    const float* g3     = (const float*)d_in[10];
    const float* b3     = (const float*)d_in[11];
    const float* w_skip = (const float*)d_in[12];
    const float* gs     = (const float*)d_in[13];
    const float* bs     = (const float*)d_in[14];
    float* out = (float*)d_out;

    // Workspace layout (floats). Total = 22,717,440 floats ~= 87 MiB.
    //   [0,2560)            BN stats  (sum/sumsq x4 stages)
    //   [2560,5120)         BN scale/bias x4 stages
    //   y1 region (3,211,264) is recycled for tblW/tblI/wdcT after build_xp.
    float* st   = (float*)d_ws;
    float* scl  = st + 2560;
    float* y1   = st + 5120;
    float* tblW = y1;                       // 903,168 floats (after y1 consumed)
    float* tblI = y1 + 903168;              // 903,168 ints
    float* wdcT = y1 + 1806336;             // 147,456 floats
    float* xp   = st + 3216384;             // 3,444,736 floats
    float* out2 = st + 6661120;             // 3,211,264 floats
    float* y3   = st + 9872384;             // 12,845,056 floats; ys lives in d_out

    float *s1 = st,       *q1 = st + 128;
    float *s2 = st + 256, *q2 = st + 384;
    float *s3 = st + 512, *q3 = st + 1024;
    float *ss = st + 1536,*qs = st + 2048;
    float *a1 = scl,        *c1 = scl + 128;
    float *a2 = scl + 256,  *c2 = scl + 384;
    float *a3 = scl + 512,  *c3 = scl + 1024;
    float *as_ = scl + 1536,*cs_ = scl + 2048;

    zero_f<<<10, 256, 0, stream>>>(st, 2560);

    // Stage 1: y1 = w1 @ x  (M=128, K=256) + stats
    gemm1x1_wmma<<<dim3(49, 1, B_), 256, 0, stream>>>(w1, x, y1, nullptr, nullptr,
                                                      P_, CIN, s1, q1);
    finalize_bn<<<1, 128, 0, stream>>>(s1, q1, g1, b1, a1, c1, P_);
    build_xp<<<13456, 256, 0, stream>>>(y1, a1, c1, xp);         // padded relu(bn(y1))

    // Offsets conv + bilinear tables (overwrites y1 region — y1 is consumed)
    offsets_tables<<<196, 128, 0, stream>>>(xp, w_off, b_off,
                                            (float4*)tblW, (int4*)tblI);
    transpose_wdc<<<576, 256, 0, stream>>>(w_dc, wdcT);

    // Deformable conv as WMMA GEMM (M=128, K=1152) + stats
    deform_wmma<<<dim3(49, 1, B_), 256, 0, stream>>>(wdcT, xp, (const float4*)tblW,
                                                     (const int4*)tblI, out2, s2, q2);
    finalize_bn<<<1, 128, 0, stream>>>(s2, q2, g2, b2, a2, c2, P_);

    // Stage 3: y3 = w3 @ relu(bn(out2))  (M=512, K=128, BN+relu fused into B load)
    gemm1x1_wmma<<<dim3(49, 4, B_), 256, 0, stream>>>(w3, out2, y3, a2, c2,
                                                      COUT, P_, s3, q3);
    // Skip: ys = w_skip @ x  (M=512, K=256) -> stored in d_out
    gemm1x1_wmma<<<dim3(49, 4, B_), 256, 0, stream>>>(w_skip, x, out, nullptr, nullptr,
                                                      COUT, CIN, ss, qs);
    finalize_bn<<<2, 256, 0, stream>>>(s3, q3, g3, b3, a3, c3, COUT);
    finalize_bn<<<2, 256, 0, stream>>>(ss, qs, gs, bs, as_, cs_, COUT);

    // out = relu(bn(y3) + bn(ys))
    final_out<<<50176, 256, 0, stream>>>(y3, out, a3, c3, as_, cs_);
}